// HierVAE_72842645340344
// MI455X (gfx1250) — compile-verified
//
#include <hip/hip_runtime.h>
#include <math.h>

typedef __attribute__((ext_vector_type(2))) float v2f;
typedef __attribute__((ext_vector_type(8))) float v8f;

#define HV_B 4096
#define HV_H 256
#define HV_L 64

// ---------------------------------------------------------------------------
// lower_bound over a sorted int array (uniform across the block)
// ---------------------------------------------------------------------------
__device__ __forceinline__ int hv_lower_bound(const int* __restrict__ a, int n, int v) {
    int lo = 0, hi = n;
    while (lo < hi) {
        int mid = (lo + hi) >> 1;
        if (a[mid] < v) lo = mid + 1; else hi = mid;
    }
    return lo;
}

// ---------------------------------------------------------------------------
// Kernel 1: segment means for tree_vecs and inter_vecs (seg sorted).
// One block per molecule b, one thread per column. Deterministic, atomic-free.
// ---------------------------------------------------------------------------
__global__ __launch_bounds__(HV_H) void hvae_segmean_kernel(
    const float* __restrict__ tree, const float* __restrict__ inter,
    const int* __restrict__ seg, int n,
    float* __restrict__ tmean, float* __restrict__ imean)
{
    const int b   = blockIdx.x;
    const int col = threadIdx.x;
    const int start = hv_lower_bound(seg, n, b);
    const int end   = hv_lower_bound(seg, n, b + 1);
    float st = 0.f, si = 0.f;
    for (int r = start; r < end; ++r) {
        st += tree [r * HV_H + col];
        si += inter[r * HV_H + col];
    }
    const float inv = 1.f / fmaxf((float)(end - start), 1.f);
    tmean[b * HV_H + col] = st * inv;
    imean[b * HV_H + col] = si * inv;
}

// ---------------------------------------------------------------------------
// Kernel 2: anchor pooling: g_pool[b] = sum over anchors in segment b of
// graph_vecs[anchor_idx[r]]. Sorted anchor_seg -> contiguous ranges.
// ---------------------------------------------------------------------------
__global__ __launch_bounds__(HV_H) void hvae_anchor_kernel(
    const float* __restrict__ gvecs, const int* __restrict__ aidx,
    const int* __restrict__ aseg, int n, float* __restrict__ gpool)
{
    const int b   = blockIdx.x;
    const int col = threadIdx.x;
    const int start = hv_lower_bound(aseg, n, b);
    const int end   = hv_lower_bound(aseg, n, b + 1);
    float s = 0.f;
    for (int r = start; r < end; ++r) {
        const int gi = aidx[r];
        s += gvecs[gi * HV_H + col];
    }
    gpool[b * HV_H + col] = s;
}

// ---------------------------------------------------------------------------
// Kernel 3: WMMA GEMM + reparameterization + KL partials.
// Each wave (32 lanes) owns a 16-row tile of B and all 64 output columns.
// Uses V_WMMA_F32_16X16X4_F32 (exact f32 precision, matches reference).
//
// f32 16x16x4 fragment layouts (ISA 7.12.2):
//   A (16x4):  lane = half*16+lm ; a[j] = A[M=lm][K = 2*half + j]
//   B (4x16):  b[j] = B[K = 2*half + j][N = lm]
//   C/D:       c[i] = C[M = i + 8*half][N = lm]
// ---------------------------------------------------------------------------
__global__ __launch_bounds__(128) void hvae_gemm_kernel(
    const float* __restrict__ root,  const float* __restrict__ tmean,
    const float* __restrict__ imean, const float* __restrict__ gpool,
    const float* __restrict__ Tmw, const float* __restrict__ Tmb,
    const float* __restrict__ Tvw, const float* __restrict__ Tvb,
    const float* __restrict__ Gmw, const float* __restrict__ Gmb,
    const float* __restrict__ Gvw, const float* __restrict__ Gvb,
    const float* __restrict__ epsT, const float* __restrict__ epsG,
    float* __restrict__ out, float* __restrict__ klPartial)
{
    const int wave = threadIdx.x >> 5;
    const int lane = threadIdx.x & 31;
    const int half = lane >> 4;
    const int lm   = lane & 15;
    const int tile = blockIdx.x * 4 + wave;     // 16-row tile index, 256 total
    const int mrow = tile * 16 + lm;            // A-row this lane loads

    v8f aTm[4], aTv[4], aGm[4], aGv[4];
    for (int nt = 0; nt < 4; ++nt) {
        const int n = nt * 16 + lm;
        const float bm = Tmb[n], bv = Tvb[n], gm = Gmb[n], gv = Gvb[n];
        for (int i = 0; i < 8; ++i) {
            aTm[nt][i] = bm; aTv[nt][i] = bv;
            aGm[nt][i] = gm; aGv[nt][i] = gv;
        }
    }

    // ---- tree branch: t_in = [root | tree_mean | inter_mean], K = 768 ----
    for (int k = 0; k < 3 * HV_H; k += 4) {
        const float* A; int kc;
        if      (k < HV_H)     { A = root;  kc = k;            }
        else if (k < 2 * HV_H) { A = tmean; kc = k - HV_H;     }
        else                   { A = imean; kc = k - 2 * HV_H; }
        v2f a;
        a.x = A[mrow * HV_H + kc + 2 * half];
        a.y = A[mrow * HV_H + kc + 2 * half + 1];
        const int kr = k + 2 * half;
        for (int nt = 0; nt < 4; ++nt) {
            const int n = nt * 16 + lm;
            v2f bm; bm.x = Tmw[kr * HV_L + n]; bm.y = Tmw[(kr + 1) * HV_L + n];
            aTm[nt] = __builtin_amdgcn_wmma_f32_16x16x4_f32(
                false, a, false, bm, (short)0, aTm[nt], false, false);
            v2f bv; bv.x = Tvw[kr * HV_L + n]; bv.y = Tvw[(kr + 1) * HV_L + n];
            aTv[nt] = __builtin_amdgcn_wmma_f32_16x16x4_f32(
                false, a, false, bv, (short)0, aTv[nt], false, false);
        }
    }

    // ---- graph branch: g_pool, K = 256 ----
    for (int k = 0; k < HV_H; k += 4) {
        v2f a;
        a.x = gpool[mrow * HV_H + k + 2 * half];
        a.y = gpool[mrow * HV_H + k + 2 * half + 1];
        const int kr = k + 2 * half;
        for (int nt = 0; nt < 4; ++nt) {
            const int n = nt * 16 + lm;
            v2f bm; bm.x = Gmw[kr * HV_L + n]; bm.y = Gmw[(kr + 1) * HV_L + n];
            aGm[nt] = __builtin_amdgcn_wmma_f32_16x16x4_f32(
                false, a, false, bm, (short)0, aGm[nt], false, false);
            v2f bv; bv.x = Gvw[kr * HV_L + n]; bv.y = Gvw[(kr + 1) * HV_L + n];
            aGv[nt] = __builtin_amdgcn_wmma_f32_16x16x4_f32(
                false, a, false, bv, (short)0, aGv[nt], false, false);
        }
    }

    // ---- epilogue: z = mean + exp(0.5*logvar)*eps ; KL partials ----
    float klp = 0.f;
    for (int nt = 0; nt < 4; ++nt) {
        const int n = nt * 16 + lm;
        for (int i = 0; i < 8; ++i) {
            const int row = tile * 16 + i + 8 * half;
            const float zm  = aTm[nt][i];
            const float zlv = -fabsf(aTv[nt][i]);
            out[row * (2 * HV_L) + n] = zm + expf(0.5f * zlv) * epsT[row * HV_L + n];
            klp += 1.f + zlv - zm * zm - expf(zlv);

            const float gm  = aGm[nt][i];
            const float glv = -fabsf(aGv[nt][i]);
            out[row * (2 * HV_L) + HV_L + n] = gm + expf(0.5f * glv) * epsG[row * HV_L + n];
            klp += 1.f + glv - gm * gm - expf(glv);
        }
    }

    __shared__ float red[128];
    red[threadIdx.x] = klp;
    __syncthreads();
    for (int s = 64; s > 0; s >>= 1) {
        if (threadIdx.x < s) red[threadIdx.x] += red[threadIdx.x + s];
        __syncthreads();
    }
    if (threadIdx.x == 0) klPartial[blockIdx.x] = red[0];
}

// ---------------------------------------------------------------------------
// Kernel 4: deterministic final KL reduction (64 partials, fixed order).
// kl = -0.5/B * sum(1 + zlv - zm^2 - exp(zlv))  over both branches.
// ---------------------------------------------------------------------------
__global__ void hvae_kl_reduce_kernel(const float* __restrict__ partials, int n,
                                      float* __restrict__ out_scalar)
{
    if (threadIdx.x == 0 && blockIdx.x == 0) {
        float s = 0.f;
        for (int i = 0; i < n; ++i) s += partials[i];
        *out_scalar = s * (-0.5f / (float)HV_B);
    }
}

// ---------------------------------------------------------------------------
extern "C" void kernel_launch(void* const* d_in, const int* in_sizes, int n_in,
                              void* d_out, int out_size, void* d_ws, size_t ws_size,
                              hipStream_t stream) {
    const float* root  = (const float*)d_in[0];
    const float* tree  = (const float*)d_in[1];
    const float* inter = (const float*)d_in[2];
    const float* gvecs = (const float*)d_in[3];
    const float* Tmw   = (const float*)d_in[4];
    const float* Tmb   = (const float*)d_in[5];
    const float* Tvw   = (const float*)d_in[6];
    const float* Tvb   = (const float*)d_in[7];
    const float* Gmw   = (const float*)d_in[8];
    const float* Gmb   = (const float*)d_in[9];
    const float* Gvw   = (const float*)d_in[10];
    const float* Gvb   = (const float*)d_in[11];
    const float* epsT  = (const float*)d_in[12];
    const float* epsG  = (const float*)d_in[13];
    const int*   tseg  = (const int*)d_in[14];
    const int*   aidx  = (const int*)d_in[15];
    const int*   aseg  = (const int*)d_in[16];

    float* out = (float*)d_out;
    float* ws  = (float*)d_ws;

    float* tmean = ws;                         // [B,H]
    float* imean = ws + 1 * HV_B * HV_H;       // [B,H]
    float* gpool = ws + 2 * HV_B * HV_H;       // [B,H]
    float* klp   = ws + 3 * HV_B * HV_H;       // [64]

    const int nTree = in_sizes[14];
    const int nAnch = in_sizes[15];

    hvae_segmean_kernel<<<HV_B, HV_H, 0, stream>>>(tree, inter, tseg, nTree, tmean, imean);
    hvae_anchor_kernel <<<HV_B, HV_H, 0, stream>>>(gvecs, aidx, aseg, nAnch, gpool);

    const int nGemmBlocks = HV_B / 64;  // 64 blocks * 4 waves * 16 rows = 4096
    hvae_gemm_kernel<<<nGemmBlocks, 128, 0, stream>>>(
        root, tmean, imean, gpool,
        Tmw, Tmb, Tvw, Tvb, Gmw, Gmb, Gvw, Gvb,
        epsT, epsG, out, klp);

    hvae_kl_reduce_kernel<<<1, 32, 0, stream>>>(klp, nGemmBlocks, out + HV_B * 2 * HV_L);
}